// dense_CNNModel_73358041416200
// MI455X (gfx1250) — compile-verified
//
#include <hip/hip_runtime.h>
#include <hip/hip_bf16.h>
#include <math.h>

// Problem constants (from reference)
#define BB_   128
#define TT_   2048
#define FF_   512
#define OS_   20
#define OF_   42
#define NC_   10
#define NCH_  16      // t-chunks for deterministic partial sums (T/NCH = 128)
#define SEGP_ 48      // padded segment width (42 -> 48, multiple of 16)
#define NSAMP (BB_*OS_)   // 2560
#define ACTLD (5*SEGP_)   // 240

typedef float v2f __attribute__((ext_vector_type(2)));
typedef float v4f __attribute__((ext_vector_type(4)));
typedef float v8f __attribute__((ext_vector_type(8)));

// ---------------------------------------------------------------------------
// K0: streaming pass over y -> chunked partial sums  partial[b][ch][f]
// Bandwidth kernel: 128 lanes x float4 = full 512-float row per iteration.
// ---------------------------------------------------------------------------
__global__ void pool_partial(const float* __restrict__ y,
                             const int* __restrict__ lengths,
                             float* __restrict__ partial) {
    int tid = threadIdx.x;            // 0..127 -> f4 column
    int ch  = blockIdx.y;             // 0..15
    int b   = blockIdx.z;             // 0..127
    int fs  = lengths[b] - (OS_ - 1);
    int t0  = ch * (TT_ / NCH_);
    int t1  = t0 + (TT_ / NCH_); if (t1 > fs) t1 = fs;
    v4f acc = {0.f, 0.f, 0.f, 0.f};
    for (int t = t0; t < t1; ++t) {
        const v4f* row = (const v4f*)(y + ((size_t)b * TT_ + t) * FF_);
        v4f v = __builtin_nontemporal_load(&row[tid]);
        acc += v;
    }
    ((v4f*)partial)[(size_t)(b * NCH_ + ch) * (FF_ / 4) + tid] = acc;
}

// ---------------------------------------------------------------------------
// K1: combine chunks + sliding-window update -> xpool[sample][f], sample=b*20+s
// win[s+1] = win[s] - y[b,s,f] + y[b,fs+s,f];  x = win/fs
// ---------------------------------------------------------------------------
__global__ void pool_finalize(const float* __restrict__ y,
                              const int* __restrict__ lengths,
                              const float* __restrict__ partial,
                              float* __restrict__ xpool) {
    int b = blockIdx.x, tid = threadIdx.x;     // 256 threads
    int fs = lengths[b] - (OS_ - 1);
    float inv = 1.f / (float)fs;
    for (int f = tid; f < FF_; f += 256) {
        float w = 0.f;
        for (int ch = 0; ch < NCH_; ++ch)
            w += partial[(size_t)(b * NCH_ + ch) * FF_ + f];
        xpool[((size_t)b * OS_ + 0) * FF_ + f] = w * inv;
        for (int s = 1; s < OS_; ++s) {
            w += y[((size_t)b * TT_ + (fs + s - 1)) * FF_ + f]
               - y[((size_t)b * TT_ + (s - 1)) * FF_ + f];
            xpool[((size_t)b * OS_ + s) * FF_ + f] = w * inv;
        }
    }
}

// ---------------------------------------------------------------------------
// K2: BN stats per channel f over 2560 samples -> folded scale/shift
// ---------------------------------------------------------------------------
__global__ void bn_stats(const float* __restrict__ xpool,
                         const float* __restrict__ gamma,
                         const float* __restrict__ beta,
                         float* __restrict__ scale,
                         float* __restrict__ shift) {
    int f = blockIdx.x, tid = threadIdx.x;     // 128 threads
    float s1 = 0.f, s2 = 0.f;
    for (int j = 0; j < NSAMP / 128; ++j) {
        float v = xpool[(size_t)(tid + j * 128) * FF_ + f];
        s1 += v; s2 += v * v;
    }
    __shared__ float r1[128], r2[128];
    r1[tid] = s1; r2[tid] = s2; __syncthreads();
    for (int off = 64; off > 0; off >>= 1) {
        if (tid < off) { r1[tid] += r1[tid + off]; r2[tid] += r2[tid + off]; }
        __syncthreads();
    }
    if (tid == 0) {
        float mean = r1[0] * (1.f / (float)NSAMP);
        float var  = r2[0] * (1.f / (float)NSAMP) - mean * mean;
        float rs   = rsqrtf(var + 1e-5f);
        float g    = gamma[f];
        scale[f] = rs * g;
        shift[f] = beta[f] - mean * rs * g;
    }
}

__global__ void bn_apply(float* __restrict__ xpool,
                         const float* __restrict__ scale,
                         const float* __restrict__ shift) {
    int i = blockIdx.x * 256 + threadIdx.x;
    if (i < NSAMP * FF_) {
        int f = i & (FF_ - 1);
        xpool[i] = xpool[i] * scale[f] + shift[f];
    }
}

// ---------------------------------------------------------------------------
// K3: repack weights W(42 x Korig) into padded B-matrix Bp(Kpad x 48),
// Bp[k][n] = W[n][seg*origSeg + kc] with zero padding (rows 42..47, kc>=origSeg)
// ---------------------------------------------------------------------------
__global__ void repack_w(const float* __restrict__ W, float* __restrict__ Bp,
                         int origSeg, int padSeg, int Korig, int total) {
    int i = blockIdx.x * 256 + threadIdx.x;
    if (i >= total) return;
    int k = i / SEGP_, n = i % SEGP_;
    int seg = k / padSeg, kc = k % padSeg;
    float v = 0.f;
    if (n < OF_ && kc < origSeg) v = W[n * Korig + seg * origSeg + kc];
    Bp[i] = v;
}

__global__ void pack_bias(const float* b1, const float* b2, const float* b3,
                          const float* b4, const float* b5, float* bp) {
    int i = threadIdx.x;
    if (i < 5 * SEGP_) {
        int layer = i / SEGP_, n = i % SEGP_;
        const float* src = layer == 0 ? b1 : layer == 1 ? b2 : layer == 2 ? b3
                         : layer == 3 ? b4 : b5;
        bp[i] = (n < OF_) ? src[n] : 0.f;
    }
}

// ---------------------------------------------------------------------------
// K4: full-precision WMMA GEMM: C(16x16 tile) = relu(A(MxK) @ Bp(Kx48) + bias)
// One wave per tile. V_WMMA_F32_16X16X4_F32, ISA 7.12.2 fragment layouts:
//  A 16x4: lanes 0-15 hold (M=lane, K=kb+{0,1}), lanes 16-31 (M=lane-16, K=kb+{2,3})
//  B 4x16: v0/v1 hold rows kb / kb+1 (lanes 0-15), kb+2 / kb+3 (lanes 16-31)
//  C: VGPR j -> rows tileM+j (lanes 0-15) and tileM+j+8 (lanes 16-31)
// ---------------------------------------------------------------------------
__global__ void wmma_gemm_relu(const float* __restrict__ A, int lda, int K,
                               const float* __restrict__ Bp,
                               const float* __restrict__ bias,
                               float* __restrict__ C, int ldc) {
    int lane  = threadIdx.x;          // 0..31 (wave32, EXEC all ones)
    int tileM = blockIdx.x * 16;
    int tileN = blockIdx.y * 16;
    int half  = lane >> 4;            // 0 or 1
    int l15   = lane & 15;
    const float* Arow = A + (size_t)(tileM + l15) * lda;
    v8f acc = {0.f, 0.f, 0.f, 0.f, 0.f, 0.f, 0.f, 0.f};
    for (int k0 = 0; k0 < K; k0 += 4) {
        int kb = k0 + half * 2;
        v2f a = *(const v2f*)(Arow + kb);
        v2f b;
        b.x = Bp[(size_t)kb * SEGP_ + tileN + l15];
        b.y = Bp[(size_t)(kb + 1) * SEGP_ + tileN + l15];
        acc = __builtin_amdgcn_wmma_f32_16x16x4_f32(
            false, a, false, b, (short)0, acc, false, false);
    }
    int col = tileN + l15;
    float bv = bias[col];
#pragma unroll
    for (int j = 0; j < 8; ++j) {
        int row = tileM + j + half * 8;
        float v = acc[j] + bv;
        C[(size_t)row * ldc + col] = v > 0.f ? v : 0.f;
    }
}

// ---------------------------------------------------------------------------
// K5: final linear (10 x 4200) + softmax per batch row.
// flat index i = c_global*20 + l ; act column = seg*48 + (c_global - seg*42)
// ---------------------------------------------------------------------------
__global__ void final_linear_softmax(const float* __restrict__ act,
                                     const float* __restrict__ Wlin,
                                     const float* __restrict__ blin,
                                     float* __restrict__ out) {
    int b = blockIdx.x, tid = threadIdx.x;     // 256 threads
    float accs[NC_];
#pragma unroll
    for (int n = 0; n < NC_; ++n) accs[n] = 0.f;
    for (int i = tid; i < 5 * OF_ * OS_; i += 256) {
        int cg = i / OS_, l = i - cg * OS_;
        int seg = cg / OF_, cc = cg - seg * OF_;
        float a = act[((size_t)b * OS_ + l) * ACTLD + seg * SEGP_ + cc];
#pragma unroll
        for (int n = 0; n < NC_; ++n) accs[n] += a * Wlin[n * (5 * OF_ * OS_) + i];
    }
    __shared__ float red[256];
    __shared__ float logits[NC_];
    for (int n = 0; n < NC_; ++n) {
        red[tid] = accs[n]; __syncthreads();
        for (int off = 128; off > 0; off >>= 1) {
            if (tid < off) red[tid] += red[tid + off];
            __syncthreads();
        }
        if (tid == 0) logits[n] = red[0] + blin[n];
        __syncthreads();
    }
    if (tid == 0) {
        float m = logits[0];
        for (int n = 1; n < NC_; ++n) m = fmaxf(m, logits[n]);
        float e[NC_], s = 0.f;
        for (int n = 0; n < NC_; ++n) { e[n] = expf(logits[n] - m); s += e[n]; }
        for (int n = 0; n < NC_; ++n) out[b * NC_ + n] = e[n] / s;
    }
}

// ---------------------------------------------------------------------------
extern "C" void kernel_launch(void* const* d_in, const int* in_sizes, int n_in,
                              void* d_out, int out_size, void* d_ws, size_t ws_size,
                              hipStream_t stream) {
    const float* y      = (const float*)d_in[0];
    const int*   len    = (const int*)  d_in[1];
    const float* gamma  = (const float*)d_in[2];
    const float* beta   = (const float*)d_in[3];
    const float* W1     = (const float*)d_in[4];
    const float* b1     = (const float*)d_in[5];
    const float* W2     = (const float*)d_in[6];
    const float* b2     = (const float*)d_in[7];
    const float* W3     = (const float*)d_in[8];
    const float* b3     = (const float*)d_in[9];
    const float* W4     = (const float*)d_in[10];
    const float* b4     = (const float*)d_in[11];
    const float* W5     = (const float*)d_in[12];
    const float* b5     = (const float*)d_in[13];
    const float* Wlin   = (const float*)d_in[14];
    const float* blin   = (const float*)d_in[15];
    float* out = (float*)d_out;

    // Workspace layout (float offsets)
    float* ws = (float*)d_ws;
    float* partial = ws;                                  // 128*16*512   = 1,048,576
    float* xpool   = partial + (size_t)BB_ * NCH_ * FF_;  // 2560*512     = 1,310,720
    float* scale   = xpool   + (size_t)NSAMP * FF_;       // 512
    float* shift   = scale + FF_;                         // 512
    float* bp1     = shift + FF_;                         // 512*48
    float* bp2     = bp1 + 512 * SEGP_;                   // 48*48
    float* bp3     = bp2 + SEGP_ * SEGP_;                 // 96*48
    float* bp4     = bp3 + 2 * SEGP_ * SEGP_;             // 144*48
    float* bp5     = bp4 + 3 * SEGP_ * SEGP_;             // 192*48
    float* biasp   = bp5 + 4 * SEGP_ * SEGP_;             // 240
    float* act     = biasp + 5 * SEGP_;                   // 2560*240

    // K0: stream y once (bandwidth-bound phase)
    pool_partial<<<dim3(1, NCH_, BB_), 128, 0, stream>>>(y, len, partial);
    // K1: sliding-window finalize -> xpool (sample-major A matrix)
    pool_finalize<<<BB_, 256, 0, stream>>>(y, len, partial, xpool);
    // K2: BN stats -> folded affine; apply in place
    bn_stats<<<FF_, 128, 0, stream>>>(xpool, gamma, beta, scale, shift);
    bn_apply<<<(NSAMP * FF_ + 255) / 256, 256, 0, stream>>>(xpool, scale, shift);
    // K3: weight repack into padded B layout
    repack_w<<<(512 * SEGP_ + 255) / 256, 256, 0, stream>>>(W1, bp1, 512, 512, 512, 512 * SEGP_);
    repack_w<<<(SEGP_ * SEGP_ + 255) / 256, 256, 0, stream>>>(W2, bp2, OF_, SEGP_, OF_, SEGP_ * SEGP_);
    repack_w<<<(2 * SEGP_ * SEGP_ + 255) / 256, 256, 0, stream>>>(W3, bp3, OF_, SEGP_, 2 * OF_, 2 * SEGP_ * SEGP_);
    repack_w<<<(3 * SEGP_ * SEGP_ + 255) / 256, 256, 0, stream>>>(W4, bp4, OF_, SEGP_, 3 * OF_, 3 * SEGP_ * SEGP_);
    repack_w<<<(4 * SEGP_ * SEGP_ + 255) / 256, 256, 0, stream>>>(W5, bp5, OF_, SEGP_, 4 * OF_, 4 * SEGP_ * SEGP_);
    pack_bias<<<1, 256, 0, stream>>>(b1, b2, b3, b4, b5, biasp);

    // K4: DenseNet pointwise convs as f32 WMMA GEMMs (concat = column ranges)
    dim3 gg(NSAMP / 16, SEGP_ / 16);   // 160 x 3 tiles, one wave each
    wmma_gemm_relu<<<gg, 32, 0, stream>>>(xpool, FF_, FF_, bp1, biasp + 0 * SEGP_, act + 0 * SEGP_, ACTLD);
    wmma_gemm_relu<<<gg, 32, 0, stream>>>(act, ACTLD,  48, bp2, biasp + 1 * SEGP_, act + 1 * SEGP_, ACTLD);
    wmma_gemm_relu<<<gg, 32, 0, stream>>>(act, ACTLD,  96, bp3, biasp + 2 * SEGP_, act + 2 * SEGP_, ACTLD);
    wmma_gemm_relu<<<gg, 32, 0, stream>>>(act, ACTLD, 144, bp4, biasp + 3 * SEGP_, act + 3 * SEGP_, ACTLD);
    wmma_gemm_relu<<<gg, 32, 0, stream>>>(act, ACTLD, 192, bp5, biasp + 4 * SEGP_, act + 4 * SEGP_, ACTLD);

    // K5: final linear + softmax
    final_linear_softmax<<<BB_, 256, 0, stream>>>(act, Wlin, blin, out);
}